// ConditinalBBP_6725918785706
// MI455X (gfx1250) — compile-verified
//
#include <hip/hip_runtime.h>
#include <math.h>

// ---------------------------------------------------------------------------
// ConditionalBBP forward loss on gfx1250 (CDNA5).
//  Pass 1 (WMMA): w_in = tanh([in_w[idx] || cov_w[cov]] @ lin_w^T + lin_b)
//                        + softplus(in_rho)*eps_in,  fused post/prior sums.
//  Pass 2: out-side gathers, dot products vs w_in, log-sigmoids, reduction.
// ---------------------------------------------------------------------------

typedef __attribute__((ext_vector_type(2))) float v2f;
typedef __attribute__((ext_vector_type(8))) float v8f;

#define EMBED   128
#define K2      256      // concat(mu_in, y) width
#define BATCH   16384
#define WIN     8
#define BW      (BATCH * WIN)

__device__ __forceinline__ float softplusf(float x) {
    // log(exp(x)+1), numerically stable
    return fmaxf(x, 0.0f) + log1pf(expf(-fabsf(x)));
}
__device__ __forceinline__ float logsigmoidf(float x) {
    // log(1/(1+exp(-x)))
    return fminf(x, 0.0f) - log1pf(expf(-fabsf(x)));
}
__device__ __forceinline__ float prior_termf(float w) {
    // log(0.5*exp(-w^2/(2*1^2)) + 0.5*exp(-w^2/(2*0.2^2)))
    float ww = w * w;
    return logf(0.5f * (expf(-0.5f * ww) + expf(-12.5f * ww)));
}
__device__ __forceinline__ float wave_sum32(float v) {
#pragma unroll
    for (int m = 16; m >= 1; m >>= 1) v += __shfl_xor(v, m, 32);
    return v;
}

__global__ void bbp_init_kernel(float* acc) { acc[0] = 0.0f; }

// ---------------------------------------------------------------------------
// Pass 1: one wave per 16 batch rows. GEMM 16x128 (K=256) via
// V_WMMA_F32_16X16X4_F32, 8 accumulators of 16x16, fused epilogue.
// grid = BATCH/(16*8) blocks of 256 threads (8 waves).
// ---------------------------------------------------------------------------
__global__ __launch_bounds__(256)
void bbp_in_side_kernel(const int* __restrict__ inputs,
                        const int* __restrict__ covars,
                        const float* __restrict__ wt,
                        const float* __restrict__ eps_in,
                        const float* __restrict__ in_w,
                        const float* __restrict__ in_rho_w,
                        const float* __restrict__ cov_w,
                        const float* __restrict__ lin_w,
                        const float* __restrict__ lin_b,
                        float* __restrict__ w_in_out,
                        float* __restrict__ acc)
{
    const int lane    = threadIdx.x & 31;
    const int wave    = threadIdx.x >> 5;
    const int rowBase = (blockIdx.x * 8 + wave) * 16;
    const int mlo     = lane & 15;   // A: row M, B: col N
    const int grp     = lane >> 4;   // K sub-group (0 -> k+0/1, 1 -> k+2/3)

    // A-matrix row gather pointers (per lane: row mlo of this tile)
    const int wordA = inputs[rowBase + mlo];
    const int covA  = covars[rowBase + mlo];
    const float* aIn  = in_w  + (size_t)wordA * EMBED;
    const float* aCov = cov_w + (size_t)covA  * EMBED;

    v8f accm[8] = {};   // 8 N-tiles of 16 -> full 128-wide output strip

    // K = 0..127 : mu_in half of the concat
#pragma unroll 2
    for (int k0 = 0; k0 < EMBED; k0 += 4) {
        const int k = k0 + grp * 2;
        v2f a = *(const v2f*)(aIn + k);
#pragma unroll
        for (int nt = 0; nt < 8; ++nt) {
            v2f b = *(const v2f*)(lin_w + (size_t)(nt * 16 + mlo) * K2 + k);
            accm[nt] = __builtin_amdgcn_wmma_f32_16x16x4_f32(
                false, a, false, b, (short)0, accm[nt], false, false);
        }
    }
    // K = 128..255 : covariate half of the concat
#pragma unroll 2
    for (int k0 = EMBED; k0 < K2; k0 += 4) {
        const int k = k0 + grp * 2;
        v2f a = *(const v2f*)(aCov + (k - EMBED));
#pragma unroll
        for (int nt = 0; nt < 8; ++nt) {
            v2f b = *(const v2f*)(lin_w + (size_t)(nt * 16 + mlo) * K2 + k);
            accm[nt] = __builtin_amdgcn_wmma_f32_16x16x4_f32(
                false, a, false, b, (short)0, accm[nt], false, false);
        }
    }

    // Epilogue: D layout -> element (M = r + 8*grp, N = nt*16 + mlo)
    const float wtv = wt[0];
    float s = 0.0f;
#pragma unroll
    for (int r = 0; r < 8; ++r) {
        const int m   = grp * 8 + r;
        const int row = rowBase + m;
        const int word = inputs[row];
        const float* rhoRow = in_rho_w + (size_t)word * EMBED;
        const float* epsRow = eps_in   + (size_t)row  * EMBED;
#pragma unroll
        for (int nt = 0; nt < 8; ++nt) {
            const int n = nt * 16 + mlo;
            float act = tanhf(accm[nt][r] + lin_b[n]);
            float sig = softplusf(rhoRow[n]);
            float eps = epsRow[n];
            float w   = act + sig * eps;
            w_in_out[(size_t)row * EMBED + n] = w;
            // post_in term minus prior_in term (all collapse into one scalar)
            s += (-0.5f * eps * eps - logf(sig)) - prior_termf(w);
        }
    }
    s = wave_sum32(s);
    if (lane == 0) {
        // in-side terms are replicated WIN times in the bw-mean
        atomicAdd(acc, wtv * s * ((float)WIN / (float)BW));
    }
}

// ---------------------------------------------------------------------------
// Pass 2: one wave per (b,w) row; lane handles 4 contiguous embed elems
// (float4 loads). grid = BW/8 blocks of 256 threads.
// ---------------------------------------------------------------------------
__global__ __launch_bounds__(256)
void bbp_out_side_kernel(const int* __restrict__ outputs,
                         const int* __restrict__ noise,
                         const float* __restrict__ wt,
                         const float* __restrict__ eps_out,
                         const float* __restrict__ out_w,
                         const float* __restrict__ out_rho_w,
                         const float* __restrict__ w_in,
                         float* __restrict__ acc)
{
    const int lane = threadIdx.x & 31;
    const int wave = threadIdx.x >> 5;
    const long r   = (long)blockIdx.x * 8 + wave;   // row in [0, BW)

    const int oi = outputs[r];
    const int ni = noise[r];

    float4 mu = ((const float4*)(out_w     + (size_t)oi      * EMBED))[lane];
    float4 rh = ((const float4*)(out_rho_w + (size_t)oi      * EMBED))[lane];
    float4 ng = ((const float4*)(out_w     + (size_t)ni      * EMBED))[lane];
    float4 ep = ((const float4*)(eps_out   + (size_t)r       * EMBED))[lane];
    float4 wi = ((const float4*)(w_in      + (size_t)(r >> 3)* EMBED))[lane];

    const float* muv = (const float*)&mu;
    const float* rhv = (const float*)&rh;
    const float* ngv = (const float*)&ng;
    const float* epv = (const float*)&ep;
    const float* wiv = (const float*)&wi;

    float post = 0.0f, prior = 0.0f, dt = 0.0f, ds = 0.0f;
#pragma unroll
    for (int j = 0; j < 4; ++j) {
        float sig = softplusf(rhv[j]);
        float wo  = muv[j] + sig * epv[j];
        post  += -0.5f * epv[j] * epv[j] - logf(sig);
        prior += prior_termf(wo);
        dt    += wiv[j] * wo;          // dot(w_in, w_out)
        ds    += wiv[j] * (-ngv[j]);   // dot(w_in, -out_w[noise])
    }
    post  = wave_sum32(post);
    prior = wave_sum32(prior);
    dt    = wave_sum32(dt);
    ds    = wave_sum32(ds);

    if (lane == 0) {
        float loss = wt[0] * (post - prior) - logsigmoidf(dt) - logsigmoidf(ds);
        atomicAdd(acc, loss * (1.0f / (float)BW));
    }
}

__global__ void bbp_fin_kernel(const float* acc, float* out) { out[0] = acc[0]; }

// ---------------------------------------------------------------------------
extern "C" void kernel_launch(void* const* d_in, const int* in_sizes, int n_in,
                              void* d_out, int out_size, void* d_ws, size_t ws_size,
                              hipStream_t stream)
{
    const int*   inputs    = (const int*)  d_in[0];
    const int*   outputs   = (const int*)  d_in[1];
    const int*   covars    = (const int*)  d_in[2];
    const int*   noise     = (const int*)  d_in[3];
    const float* wt        = (const float*)d_in[4];
    const float* eps_in    = (const float*)d_in[5];
    const float* eps_out   = (const float*)d_in[6];
    const float* in_w      = (const float*)d_in[7];
    const float* out_w     = (const float*)d_in[8];
    const float* in_rho_w  = (const float*)d_in[9];
    const float* out_rho_w = (const float*)d_in[10];
    const float* cov_w     = (const float*)d_in[11];
    const float* lin_w     = (const float*)d_in[12];
    const float* lin_b     = (const float*)d_in[13];

    float* acc      = (float*)d_ws;        // [0]: scalar accumulator
    float* w_in_buf = (float*)d_ws + 64;   // 256B-aligned: BATCH*EMBED floats (8 MB)

    bbp_init_kernel<<<dim3(1), dim3(1), 0, stream>>>(acc);

    bbp_in_side_kernel<<<dim3(BATCH / (16 * 8)), dim3(256), 0, stream>>>(
        inputs, covars, wt, eps_in, in_w, in_rho_w, cov_w, lin_w, lin_b,
        w_in_buf, acc);

    bbp_out_side_kernel<<<dim3(BW / 8), dim3(256), 0, stream>>>(
        outputs, noise, wt, eps_out, out_w, out_rho_w, w_in_buf, acc);

    bbp_fin_kernel<<<dim3(1), dim3(1), 0, stream>>>(acc, (float*)d_out);
}